// SGCNet_58016418234783
// MI455X (gfx1250) — compile-verified
//
#include <hip/hip_runtime.h>
#include <hip/hip_bf16.h>

#define NNODES 100000
#define NEDGES 1600000
#define DIN    128
#define DH     128
#define DOUT   64
#define BN_EPS 1e-5f

typedef __attribute__((ext_vector_type(2))) float v2f;
typedef __attribute__((ext_vector_type(8))) float v8f;

// ---------------- degree / norm ----------------

__global__ void degree_kernel(const int* __restrict__ row, float* __restrict__ deg, int nE) {
    int e = blockIdx.x * blockDim.x + threadIdx.x;
    if (e < nE) atomicAdd(&deg[row[e]], 1.0f);
}

__global__ void dis_kernel(const float* __restrict__ deg, float* __restrict__ dis, int n) {
    int i = blockIdx.x * blockDim.x + threadIdx.x;
    if (i < n) {
        float d = deg[i];
        dis[i] = (d > 0.0f) ? rsqrtf(fmaxf(d, 1.0f)) : 0.0f;
    }
}

__global__ void norm_kernel(const float* __restrict__ dis, const int* __restrict__ row,
                            const int* __restrict__ col, float* __restrict__ nrm, int nE) {
    int e = blockIdx.x * blockDim.x + threadIdx.x;
    if (e < nE) nrm[e] = dis[row[e]] * dis[col[e]];
}

// ---------------- scatter propagation: hout[col] += norm * hin[row] ----------------
// one thread per (edge, 4-feature chunk): 32 threads cover the 128 features of one edge

__global__ __launch_bounds__(256) void scatter_prop_kernel(
        const float* __restrict__ hin, float* __restrict__ hout,
        const int* __restrict__ row, const int* __restrict__ col,
        const float* __restrict__ nrm, int nE) {
    unsigned t = blockIdx.x * blockDim.x + threadIdx.x;   // < E*32 = 51.2M, fits u32
    unsigned e = t >> 5;
    if (e >= (unsigned)nE) return;
    unsigned fc = (t & 31u) << 2;                         // feature offset 0..124
    int rs = row[e], cd = col[e];
    float nv = nrm[e];
    const float4 s = *(const float4*)(hin + ((long long)rs << 7) + fc);
    float* dst = hout + ((long long)cd << 7) + fc;
    atomicAdd(dst + 0, nv * s.x);
    atomicAdd(dst + 1, nv * s.y);
    atomicAdd(dst + 2, nv * s.z);
    atomicAdd(dst + 3, nv * s.w);
}

// ---------------- WMMA f32 GEMM:  Y[N,FOUT] = H[N,128] @ W[128,FOUT] + b ----------------
// grid.x = N/16 (one 16-row tile per block); blockDim.x = 32 * (FOUT/16), one wave per
// 16-column output tile. K swept in steps of 4 via V_WMMA_F32_16X16X4_F32.
// FOUT is a compile-time constant so all W/Y addresses fold into immediate offsets.
// ISA layouts (CDNA5 ISA §7.12.2):
//   A 16x4 f32 : lane m=lane%16, kpair = (lane<16)?{0,1}:{2,3} -> v2f
//   B 4x16 f32 : lane n=lane%16, same kpair selection          -> v2f
//   C/D 16x16  : vgpr v holds row v (lanes 0-15) / row 8+v (lanes 16-31), col = lane%16

template<int FOUT>
__global__ __launch_bounds__(256) void wmma_gemm_bias_kernel(
        const float* __restrict__ H, const float* __restrict__ W,
        const float* __restrict__ bias, float* __restrict__ Y) {
    __shared__ float As[16 * 129];                        // stride 129: conflict-free
    const int row0 = blockIdx.x << 4;
    const int tid  = threadIdx.x;

    // stage 16x128 A tile (coalesced: consecutive tid -> consecutive k)
    for (int i = tid; i < 16 * 128; i += blockDim.x) {
        int r = i >> 7, c = i & 127;
        As[r * 129 + c] = H[((long long)(row0 + r) << 7) + c];
    }
    __syncthreads();

    const int wave = tid >> 5;
    const int lane = tid & 31;
    const int m    = lane & 15;
    const int kb   = (lane < 16) ? 0 : 2;
    const int ncol = (wave << 4) + m;

    // per-lane bases; loop offsets become compile-time immediates
    const float* Wp = W + kb * FOUT + ncol;               // +k*FOUT / +(k+1)*FOUT imm
    const float* Ap = As + m * 129 + kb;                  // +k imm (via ds offsets)

    v8f acc = {};
#pragma unroll
    for (int k = 0; k < 128; k += 4) {
        v2f a, b;
        a.x = Ap[k];
        a.y = Ap[k + 1];
        b.x = Wp[k * FOUT];
        b.y = Wp[(k + 1) * FOUT];
        acc = __builtin_amdgcn_wmma_f32_16x16x4_f32(
                /*neg_a=*/false, a, /*neg_b=*/false, b,
                /*c_mod=*/(short)0, acc, /*reuse_a=*/false, /*reuse_b=*/false);
    }

    const float bv = bias[ncol];
    const int rbase = row0 + ((lane < 16) ? 0 : 8);
    float* Yp = Y + (long long)rbase * FOUT + ncol;       // +v*FOUT imm
#pragma unroll
    for (int v = 0; v < 8; ++v)
        Yp[v * FOUT] = acc[v] + bv;
}

// ---------------- BatchNorm (training mode, biased variance) ----------------

__global__ void bn_stats_kernel(const float* __restrict__ h, float* __restrict__ sums,
                                float* __restrict__ sumsq, int n) {
    int f = threadIdx.x;                                  // blockDim.x == 128
    float s = 0.0f, s2 = 0.0f;
    for (int r = blockIdx.x; r < n; r += gridDim.x) {
        float v = h[((long long)r << 7) + f];
        s += v; s2 += v * v;
    }
    atomicAdd(&sums[f], s);
    atomicAdd(&sumsq[f], s2);
}

__global__ void bn_apply_kernel(float* __restrict__ h, const float* __restrict__ sums,
                                const float* __restrict__ sumsq,
                                const float* __restrict__ gamma,
                                const float* __restrict__ beta, int n) {
    int idx = blockIdx.x * blockDim.x + threadIdx.x;      // n*128 = 12.8M fits int
    if (idx >= n * DH) return;
    int f = idx & 127;
    float inv_n = 1.0f / (float)n;
    float mu  = sums[f] * inv_n;
    float var = sumsq[f] * inv_n - mu * mu;               // biased, == mean((x-mu)^2)
    float inv = rsqrtf(var + BN_EPS);
    h[idx] = gamma[f] * (h[idx] - mu) * inv + beta[f];
}

// ---------------- launcher ----------------

extern "C" void kernel_launch(void* const* d_in, const int* in_sizes, int n_in,
                              void* d_out, int out_size, void* d_ws, size_t ws_size,
                              hipStream_t stream) {
    const float* x   = (const float*)d_in[0];
    const int*   ei  = (const int*)d_in[1];
    const int*   row = ei;
    const int*   col = ei + NEDGES;
    const float* W1  = (const float*)d_in[2];
    const float* b1  = (const float*)d_in[3];
    const float* g1  = (const float*)d_in[4];
    const float* be1 = (const float*)d_in[5];
    const float* W2  = (const float*)d_in[6];
    const float* b2  = (const float*)d_in[7];
    const float* g2  = (const float*)d_in[8];
    const float* be2 = (const float*)d_in[9];
    const float* W3  = (const float*)d_in[10];
    const float* b3  = (const float*)d_in[11];
    float* out = (float*)d_out;

    // workspace layout (floats)
    float* ws    = (float*)d_ws;
    float* deg   = ws;                                  // N
    float* dis   = deg + NNODES;                        // N
    float* nrm   = dis + NNODES;                        // E
    float* hA    = nrm + NEDGES;                        // N*128
    float* hB    = hA + (size_t)NNODES * DH;            // N*128
    float* sums  = hB + (size_t)NNODES * DH;            // 128
    float* sumsq = sums + DH;                           // 128

    const int TPB = 256;
    const int eBlocks = (NEDGES + TPB - 1) / TPB;
    const int nBlocks = (NNODES + TPB - 1) / TPB;
    const unsigned propThreads = (unsigned)NEDGES * 32u;
    const int propBlocks = (int)((propThreads + TPB - 1) / TPB);

    // edge normalization
    hipMemsetAsync(deg, 0, NNODES * sizeof(float), stream);
    degree_kernel<<<eBlocks, TPB, 0, stream>>>(row, deg, NEDGES);
    dis_kernel<<<nBlocks, TPB, 0, stream>>>(deg, dis, NNODES);
    norm_kernel<<<eBlocks, TPB, 0, stream>>>(dis, row, col, nrm, NEDGES);

    // ---- propagate x twice: x -> hA -> hB
    hipMemsetAsync(hA, 0, (size_t)NNODES * DH * sizeof(float), stream);
    scatter_prop_kernel<<<propBlocks, TPB, 0, stream>>>(x, hA, row, col, nrm, NEDGES);
    hipMemsetAsync(hB, 0, (size_t)NNODES * DH * sizeof(float), stream);
    scatter_prop_kernel<<<propBlocks, TPB, 0, stream>>>(hA, hB, row, col, nrm, NEDGES);

    // ---- layer 1: hA = BN(hB @ W1 + b1)
    wmma_gemm_bias_kernel<DH><<<NNODES / 16, 32 * (DH / 16), 0, stream>>>(hB, W1, b1, hA);
    hipMemsetAsync(sums, 0, 2 * DH * sizeof(float), stream);
    bn_stats_kernel<<<1024, DH, 0, stream>>>(hA, sums, sumsq, NNODES);
    bn_apply_kernel<<<(NNODES * DH + TPB - 1) / TPB, TPB, 0, stream>>>(hA, sums, sumsq, g1, be1, NNODES);

    // ---- propagate twice: hA -> hB -> hA
    hipMemsetAsync(hB, 0, (size_t)NNODES * DH * sizeof(float), stream);
    scatter_prop_kernel<<<propBlocks, TPB, 0, stream>>>(hA, hB, row, col, nrm, NEDGES);
    hipMemsetAsync(hA, 0, (size_t)NNODES * DH * sizeof(float), stream);
    scatter_prop_kernel<<<propBlocks, TPB, 0, stream>>>(hB, hA, row, col, nrm, NEDGES);

    // ---- layer 2: hB = BN(hA @ W2 + b2)
    wmma_gemm_bias_kernel<DH><<<NNODES / 16, 32 * (DH / 16), 0, stream>>>(hA, W2, b2, hB);
    hipMemsetAsync(sums, 0, 2 * DH * sizeof(float), stream);
    bn_stats_kernel<<<1024, DH, 0, stream>>>(hB, sums, sumsq, NNODES);
    bn_apply_kernel<<<(NNODES * DH + TPB - 1) / TPB, TPB, 0, stream>>>(hB, sums, sumsq, g2, be2, NNODES);

    // ---- output layer: out = hB @ W3 + b3  [N, 64]
    wmma_gemm_bias_kernel<DOUT><<<NNODES / 16, 32 * (DOUT / 16), 0, stream>>>(hB, W3, b3, out);
}